// MultiHeadAttentionRel_14989435863464
// MI455X (gfx1250) — compile-verified
//
#include <hip/hip_runtime.h>
#include <hip/hip_bf16.h>
#include <cstdint>

// Problem constants
#define BB 2
#define LL 384
#define DD 256
#define HB 8
#define DHD 32
#define KT 48                 // pos rows per LDS tile
#define NT (LL / KT)          // 8 tiles
#define PSTR 260              // padded LDS row stride in floats (bank-conflict-free b64 reads)
#define TILE_F (KT * PSTR)    // floats per tile buffer
#define WROWS 16              // w_lds rows (8 heads + 8 zero rows for 16x16 WMMA)
#define INV_SQRT_DH 0.17677669529663687f

typedef float v2f  __attribute__((ext_vector_type(2)));
typedef float v8f  __attribute__((ext_vector_type(8)));
typedef unsigned int v4u __attribute__((ext_vector_type(4)));
typedef int  v4i  __attribute__((ext_vector_type(4)));
typedef int  v8i  __attribute__((ext_vector_type(8)));

#if defined(__has_builtin)
#if __has_builtin(__builtin_amdgcn_tensor_load_to_lds)
#define USE_TDM 1
#endif
#endif

__device__ __forceinline__ void wait_tensorcnt0() {
#if defined(__has_builtin) && __has_builtin(__builtin_amdgcn_s_wait_tensorcnt)
    __builtin_amdgcn_s_wait_tensorcnt(0);
#else
    asm volatile("s_wait_tensorcnt 0x0" ::: "memory");
#endif
}

#ifdef USE_TDM
// Issue one TDM 2D tile load: KT rows x 256 f32, row stride 256 f32 in memory,
// LDS padding 4 dwords after every 256 dwords -> 260-float LDS row stride.
__device__ __forceinline__ void tdm_issue_tile(const float* gsrc, unsigned lds_byte_off) {
    uint64_t ga = (uint64_t)(uintptr_t)gsrc;
    v4u g0;
    g0.x = 1u;                                             // count=1, user descriptor
    g0.y = lds_byte_off;                                   // lds_addr
    g0.z = (unsigned)(ga & 0xffffffffu);                   // global_addr[31:0]
    g0.w = (unsigned)((ga >> 32) & 0x01ffffffu)            // global_addr[56:32]
         | 0x80000000u;                                    // type=2 ("image")
    v8i g1;
    g1[0] = (2 << 16)      // data_size = 4 bytes
          | (1 << 20)      // pad_enable
          | (7 << 22)      // pad_interval: 256 dwords
          | (3 << 25);     // pad_amount: 4 dwords
    g1[1] = (int)(DD << 16);          // tensor_dim0[15:0] in bits[31:16]
    g1[2] = (int)(LL << 16);          // tensor_dim1[15:0] in bits[31:16] (dim0 hi = 0)
    g1[3] = (int)(DD << 16);          // tile_dim0 in bits[31:16] (dim1 hi = 0)
    g1[4] = KT;                       // tile_dim1 (tile_dim2 = 0)
    g1[5] = DD;                       // tensor_dim0_stride low 32 (256 elements)
    g1[6] = 0;
    g1[7] = 0;
    v4i gz4 = {0, 0, 0, 0};
    v8i gz8 = {0, 0, 0, 0, 0, 0, 0, 0};
    __builtin_amdgcn_tensor_load_to_lds(g0, g1, gz4, gz4, gz8, 0);
}
#endif

// ---------------------------------------------------------------------------
// Kernel 1: Q/K/V projections.  one block per (b,l) row, 256 threads.
// ---------------------------------------------------------------------------
__global__ void proj_qkv(const float* __restrict__ key, const float* __restrict__ query,
                         const float* __restrict__ value,
                         const float* __restrict__ Wk, const float* __restrict__ bk,
                         const float* __restrict__ Wq, const float* __restrict__ bq,
                         const float* __restrict__ Wv, const float* __restrict__ bv,
                         float* __restrict__ Qp, float* __restrict__ Kp,
                         float* __restrict__ Vp) {
    __shared__ float xq[DD], xk[DD], xv[DD];
    const int row = blockIdx.x;          // b*L + l
    const int d   = threadIdx.x;
    xq[d] = query[(size_t)row * DD + d];
    xk[d] = key  [(size_t)row * DD + d];
    xv[d] = value[(size_t)row * DD + d];
    __syncthreads();
    float aq = bq[d], ak = bk[d], av = bv[d];
    const float* wqr = Wq + (size_t)d * DD;
    const float* wkr = Wk + (size_t)d * DD;
    const float* wvr = Wv + (size_t)d * DD;
    #pragma unroll 4
    for (int j = 0; j < DD; ++j) {
        aq = fmaf(xq[j], wqr[j], aq);
        ak = fmaf(xk[j], wkr[j], ak);
        av = fmaf(xv[j], wvr[j], av);
    }
    Qp[(size_t)row * DD + d] = aq;
    Kp[(size_t)row * DD + d] = ak;
    Vp[(size_t)row * DD + d] = av;
}

// ---------------------------------------------------------------------------
// Kernel 2: build qu = Q+u, w[b,q,h,:] = (Q+v)_h @ Wr_h, c = (Q+v)_h . br_h
// ---------------------------------------------------------------------------
__global__ void make_w(const float* __restrict__ Qp, const float* __restrict__ Wr,
                       const float* __restrict__ br, const float* __restrict__ u,
                       const float* __restrict__ vparam,
                       float* __restrict__ qu, float* __restrict__ wmat,
                       float* __restrict__ cvec) {
    __shared__ float qv[DD];
    const int row = blockIdx.x;          // b*L + q
    const int t   = threadIdx.x;
    const float qval = Qp[(size_t)row * DD + t];
    qv[t] = qval + vparam[t];            // (h,dh) flattened == d
    qu[(size_t)row * DD + t] = qval + u[t];
    __syncthreads();
    #pragma unroll
    for (int h = 0; h < HB; ++h) {
        float acc = 0.f;
        #pragma unroll 4
        for (int dh = 0; dh < DHD; ++dh)
            acc = fmaf(qv[h * DHD + dh], Wr[(size_t)(h * DHD + dh) * DD + t], acc);
        wmat[(size_t)row * (HB * DD) + h * DD + t] = acc;
    }
    if (t < HB) {
        float acc = 0.f;
        for (int dh = 0; dh < DHD; ++dh)
            acc = fmaf(qv[t * DHD + dh], br[t * DHD + dh], acc);
        cvec[(size_t)row * HB + t] = acc;
    }
}

// ---------------------------------------------------------------------------
// Kernel 3: fused relative attention.  One block per (b,q): 256 threads/8 waves.
// Streams pos[b,q,:,:] via TDM double buffer; B_D via v_wmma_f32_16x16x4_f32.
// ---------------------------------------------------------------------------
__global__ void attn_rel(const float* __restrict__ pos, const float* __restrict__ kmask,
                         const float* __restrict__ Kp, const float* __restrict__ Vp,
                         const float* __restrict__ qu, const float* __restrict__ wmat,
                         const float* __restrict__ cvec, float* __restrict__ out) {
    extern __shared__ float smem[];
    float* posbuf = smem;                          // 2*TILE_F
    float* wlds   = smem + 2 * TILE_F;             // WROWS*PSTR (rows 8..15 zero)
    float* score  = wlds + WROWS * PSTR;           // HB*LL
    float* qulds  = score + HB * LL;               // DD

    const int row  = blockIdx.x;                   // b*L + q
    const int b    = row / LL;
    const int tid  = threadIdx.x;
    const int lane = tid & 31;
    const unsigned wid = (unsigned)__builtin_amdgcn_readfirstlane((int)(threadIdx.x >> 5));

#ifdef USE_TDM
    if (wid == 0) {
        tdm_issue_tile(pos + ((size_t)row * LL) * DD,
                       (unsigned)(uintptr_t)posbuf);
    }
#endif

    // Populate w_lds (16 rows x PSTR; rows >= 8 and pad columns zeroed) + qu row.
    for (int i = tid; i < WROWS * PSTR; i += 256) {
        const int r = i / PSTR;
        const int c = i - r * PSTR;
        wlds[i] = (r < HB && c < DD)
                    ? wmat[(size_t)row * (HB * DD) + r * DD + c] : 0.f;
    }
    qulds[tid] = qu[(size_t)row * DD + tid];
    __syncthreads();

    // A_C + c into score (plain stores; each (h,k) owned by one thread).
    #pragma unroll
    for (int i = 0; i < (HB * LL) / 256; ++i) {
        const int p = tid + 256 * i;               // 0..3071
        const int h = p / LL;
        const int k = p - h * LL;
        const float* kr  = Kp + ((size_t)b * LL + k) * DD + h * DHD;
        const float* quh = qulds + h * DHD;
        float acc = cvec[(size_t)row * HB + h];
        #pragma unroll 4
        for (int dh = 0; dh < DHD; ++dh) acc = fmaf(quh[dh], kr[dh], acc);
        score[h * LL + k] = acc;
    }

    // Hoisted A-matrix registers: wave 'wid' owns j in [32*wid, 32*wid+32).
    const int m  = lane & 15;        // WMMA row (head; rows 8..15 are zero)
    const int tp = lane >> 4;        // K-pair selector per 16x16x4 f32 layout
    const int jb = (tid >> 5) * 32;
    v2f areg[8];
    #pragma unroll
    for (int jj = 0; jj < 8; ++jj)
        areg[jj] = *(const v2f*)(wlds + m * PSTR + jb + jj * 4 + tp * 2);
    __syncthreads();   // score init + w_lds stable before atomics / reuse

    for (int t = 0; t < NT; ++t) {
#ifdef USE_TDM
        if (wid == 0) wait_tensorcnt0();
        __syncthreads();                            // tile t visible to all waves
        if (wid == 0 && (t + 1) < NT) {
            tdm_issue_tile(pos + ((size_t)row * LL + (size_t)(t + 1) * KT) * DD,
                           (unsigned)(uintptr_t)(posbuf + ((t + 1) & 1) * TILE_F));
        }
#else
        {   // cooperative copy fallback
            const float* src = pos + ((size_t)row * LL + (size_t)t * KT) * DD;
            float* dst = posbuf + (t & 1) * TILE_F;
            for (int i = tid * 4; i < KT * DD; i += 256 * 4) {
                const int r = i >> 8;
                const int c = i & 255;
                *(float4*)(dst + r * PSTR + c) = *(const float4*)(src + i);
            }
        }
        __syncthreads();
#endif
        const float* buf = posbuf + (t & 1) * TILE_F;
        v8f cacc[3];
        #pragma unroll
        for (int ct = 0; ct < 3; ++ct) cacc[ct] = v8f{};
        #pragma unroll
        for (int ct = 0; ct < 3; ++ct) {
            const float* bp = buf + (ct * 16 + m) * PSTR + jb + tp * 2;
            #pragma unroll
            for (int jj = 0; jj < 8; ++jj) {
                const v2f breg = *(const v2f*)(bp + jj * 4);
                cacc[ct] = __builtin_amdgcn_wmma_f32_16x16x4_f32(
                    false, areg[jj], false, breg, (short)0, cacc[ct], false, false);
            }
        }
        // lanes 0..15 hold head rows 0..7 (vgpr r, col = lane)
        if (lane < 16) {
            const int kg = t * KT + lane;
            #pragma unroll
            for (int ct = 0; ct < 3; ++ct) {
                #pragma unroll
                for (int r = 0; r < HB; ++r)
                    atomicAdd(&score[r * LL + kg + ct * 16], cacc[ct][r]);
            }
        }
        __syncthreads();
    }

    // Softmax: wave h owns head h; 12 k's per lane.
    const int h = tid >> 5;
    float sv[LL / 32];
    float smax = -3.0e38f;
    #pragma unroll
    for (int i = 0; i < LL / 32; ++i) {
        const int k = lane + 32 * i;
        const float s = score[h * LL + k] * INV_SQRT_DH
                      - (1.0f - kmask[(size_t)b * LL + k]) * 1e15f;
        sv[i] = s;
        smax = fmaxf(smax, s);
    }
    #pragma unroll
    for (int off = 16; off > 0; off >>= 1)
        smax = fmaxf(smax, __shfl_xor(smax, off, 32));
    float ssum = 0.f;
    #pragma unroll
    for (int i = 0; i < LL / 32; ++i) {
        const float e = __expf(sv[i] - smax);
        sv[i] = e;
        ssum += e;
    }
    #pragma unroll
    for (int off = 16; off > 0; off >>= 1)
        ssum += __shfl_xor(ssum, off, 32);
    const float inv = 1.0f / ssum;
    #pragma unroll
    for (int i = 0; i < LL / 32; ++i)
        score[h * LL + lane + 32 * i] = sv[i] * inv;
    __syncthreads();

    // out[b,q,h,dh] = sum_k attn[h,k] * V[b,k,h*32+dh]  (V is L2-resident)
    const int dh = tid & 31;
    const float* vbase = Vp + (size_t)b * LL * DD + h * DHD + dh;
    float acc = 0.f;
    #pragma unroll 4
    for (int k = 0; k < LL; ++k)
        acc = fmaf(score[h * LL + k], vbase[(size_t)k * DD], acc);
    out[(size_t)row * DD + h * DHD + dh] = acc;
}

// ---------------------------------------------------------------------------
extern "C" void kernel_launch(void* const* d_in, const int* in_sizes, int n_in,
                              void* d_out, int out_size, void* d_ws, size_t ws_size,
                              hipStream_t stream) {
    const float* key    = (const float*)d_in[0];
    const float* query  = (const float*)d_in[1];
    const float* value  = (const float*)d_in[2];
    const float* pos    = (const float*)d_in[3];
    const float* kmask  = (const float*)d_in[4];
    const float* Wk     = (const float*)d_in[5];
    const float* bk     = (const float*)d_in[6];
    const float* Wq     = (const float*)d_in[7];
    const float* bq     = (const float*)d_in[8];
    const float* Wv     = (const float*)d_in[9];
    const float* bv     = (const float*)d_in[10];
    const float* Wr     = (const float*)d_in[11];
    const float* br     = (const float*)d_in[12];
    const float* u      = (const float*)d_in[13];
    const float* vparam = (const float*)d_in[14];
    float* out = (float*)d_out;

    // Workspace layout (floats)
    float* ws = (float*)d_ws;
    const size_t nrow = (size_t)BB * LL;
    float* Qp   = ws;                       // nrow*DD
    float* Kp   = Qp + nrow * DD;
    float* Vp   = Kp + nrow * DD;
    float* qu   = Vp + nrow * DD;
    float* wmat = qu + nrow * DD;           // nrow*HB*DD
    float* cvec = wmat + nrow * HB * DD;    // nrow*HB

    proj_qkv<<<dim3(BB * LL), dim3(DD), 0, stream>>>(
        key, query, value, Wk, bk, Wq, bq, Wv, bv, Qp, Kp, Vp);
    make_w<<<dim3(BB * LL), dim3(DD), 0, stream>>>(
        Qp, Wr, br, u, vparam, qu, wmat, cvec);

    const size_t smem_bytes =
        (size_t)(2 * TILE_F + WROWS * PSTR + HB * LL + DD) * sizeof(float);
    attn_rel<<<dim3(BB * LL), dim3(256), smem_bytes, stream>>>(
        pos, kmask, Kp, Vp, qu, wmat, cvec, out);
}